// GAE_28587302322911
// MI455X (gfx1250) — compile-verified
//
#include <hip/hip_runtime.h>

// GAE backward scan, B=8192 rows x T=2048 steps.
// HBM-bound: ~436 MB traffic -> ~19 us at 23.3 TB/s. Strategy:
//  - 1 workgroup per row, 256 threads (8 wave32 waves), 8 consecutive t per thread
//  - fp32 streams: coalesced non-temporal global_load_b128 / global_store_b128
//  - bool streams: CDNA5 async DMA global_load_async_to_lds_b64 (ASYNCcnt path)
//  - affine-composite Hillis-Steele scan over 256 segments in LDS for the carry

#define GAMMA 0.99f
#define LMBDA 0.95f
#define T_DIM 2048
#define BLK   256
#define PER   8          // T_DIM / BLK

typedef float         v4f __attribute__((ext_vector_type(4)));
typedef unsigned int  v2u __attribute__((ext_vector_type(2)));

// Low 32 bits of a flat shared-aperture pointer == LDS byte offset
// (ISA 10.2: LDS address is the low 32 bits; HW adds the wave's LDS_BASE).
__device__ __forceinline__ unsigned lds_off(const void* p) {
    return (unsigned)(unsigned long long)p;
}

__global__ __launch_bounds__(BLK) void gae_scan_kernel(
    const float* __restrict__ reward,
    const unsigned char* __restrict__ terminated,
    const unsigned char* __restrict__ done,
    const float* __restrict__ value,
    const float* __restrict__ next_value,
    const float* __restrict__ discount,
    float* __restrict__ adv_out,
    float* __restrict__ ret_out)
{
    __shared__ __align__(16) unsigned char s_term[T_DIM];
    __shared__ __align__(16) unsigned char s_done[T_DIM];
    __shared__ float sC[BLK];
    __shared__ float sD[BLK];

    const int row  = blockIdx.x;
    const int tid  = threadIdx.x;
    const int base = row * T_DIM + tid * PER;   // fits in int: < 2^24 elements

    // ---- CDNA5 async DMA: stream bool rows into LDS (8 B per lane) ----
    {
        unsigned long long gterm = (unsigned long long)(terminated + base);
        unsigned long long gdone = (unsigned long long)(done + base);
        unsigned lterm = lds_off(&s_term[tid * PER]);
        unsigned ldone = lds_off(&s_done[tid * PER]);
        asm volatile("global_load_async_to_lds_b64 %0, %1, off"
                     :: "v"(lterm), "v"(gterm) : "memory");
        asm volatile("global_load_async_to_lds_b64 %0, %1, off"
                     :: "v"(ldone), "v"(gdone) : "memory");
    }

    // ---- coalesced, non-temporal wide loads (32B fp32 per thread per array) ----
    v4f r0 = __builtin_nontemporal_load((const v4f*)(reward     + base));
    v4f r1 = __builtin_nontemporal_load((const v4f*)(reward     + base) + 1);
    v4f v0 = __builtin_nontemporal_load((const v4f*)(value      + base));
    v4f v1 = __builtin_nontemporal_load((const v4f*)(value      + base) + 1);
    v4f n0 = __builtin_nontemporal_load((const v4f*)(next_value + base));
    v4f n1 = __builtin_nontemporal_load((const v4f*)(next_value + base) + 1);
    v4f d0 = __builtin_nontemporal_load((const v4f*)(discount   + base));
    v4f d1 = __builtin_nontemporal_load((const v4f*)(discount   + base) + 1);

    float r[PER]  = {r0.x, r0.y, r0.z, r0.w, r1.x, r1.y, r1.z, r1.w};
    float v[PER]  = {v0.x, v0.y, v0.z, v0.w, v1.x, v1.y, v1.z, v1.w};
    float nv[PER] = {n0.x, n0.y, n0.z, n0.w, n1.x, n1.y, n1.z, n1.w};
    float dc[PER] = {d0.x, d0.y, d0.z, d0.w, d1.x, d1.y, d1.z, d1.w};

    // ---- consume DMA'd bools from LDS ----
    asm volatile("s_wait_asynccnt 0x0" ::: "memory");
    v2u tb = *(const v2u*)&s_term[tid * PER];
    v2u db = *(const v2u*)&s_done[tid * PER];

    float delta[PER], coef[PER];
#pragma unroll
    for (int k = 0; k < PER; ++k) {
        unsigned int tw = (k < 4) ? tb.x : tb.y;
        unsigned int dw = (k < 4) ? db.x : db.y;
        float nonterm = 1.0f - (float)((tw >> (8 * (k & 3))) & 0xFFu);
        float nondone = 1.0f - (float)((dw >> (8 * (k & 3))) & 0xFFu);
        float g = dc[k] * GAMMA;
        delta[k] = fmaf(g * nv[k], nonterm, r[k]) - v[k];
        coef[k]  = g * LMBDA * nondone;
    }

    // ---- fold 8 elements (backward in time) into affine composite g -> C*g + D ----
    float C = coef[PER - 1];
    float D = delta[PER - 1];
#pragma unroll
    for (int k = PER - 2; k >= 0; --k) {
        D = fmaf(coef[k], D, delta[k]);
        C = coef[k] * C;
    }

    // ---- reverse exclusive scan of composites across the 256 segments ----
    // scan index j runs forward over reversed segments; combine(new,old) = new ∘ old
    const int j = (BLK - 1) - tid;
    sC[j] = C;
    sD[j] = D;
    __syncthreads();
#pragma unroll
    for (int dstep = 1; dstep < BLK; dstep <<= 1) {
        float cj = sC[j], dj = sD[j];
        float cp = 0.0f, dp = 0.0f;
        const bool has = (j >= dstep);
        if (has) { cp = sC[j - dstep]; dp = sD[j - dstep]; }
        __syncthreads();
        if (has) {
            sC[j] = cj * cp;
            sD[j] = fmaf(cj, dp, dj);
        }
        __syncthreads();
    }
    // incoming carry for this segment = exclusive composite applied to gae=0
    float gae = (j == 0) ? 0.0f : sD[j - 1];

    // ---- replay local elements with the carry, emit adv and returns ----
    float adv[PER];
#pragma unroll
    for (int k = PER - 1; k >= 0; --k) {
        gae = fmaf(coef[k], gae, delta[k]);
        adv[k] = gae;
    }

    v4f a0 = {adv[0], adv[1], adv[2], adv[3]};
    v4f a1 = {adv[4], adv[5], adv[6], adv[7]};
    v4f q0 = {adv[0] + v[0], adv[1] + v[1], adv[2] + v[2], adv[3] + v[3]};
    v4f q1 = {adv[4] + v[4], adv[5] + v[5], adv[6] + v[6], adv[7] + v[7]};

    __builtin_nontemporal_store(a0, (v4f*)(adv_out + base));
    __builtin_nontemporal_store(a1, (v4f*)(adv_out + base) + 1);
    __builtin_nontemporal_store(q0, (v4f*)(ret_out + base));
    __builtin_nontemporal_store(q1, (v4f*)(ret_out + base) + 1);
}

extern "C" void kernel_launch(void* const* d_in, const int* in_sizes, int n_in,
                              void* d_out, int out_size, void* d_ws, size_t ws_size,
                              hipStream_t stream) {
    const float*         reward     = (const float*)d_in[0];
    const unsigned char* terminated = (const unsigned char*)d_in[1];
    const unsigned char* done       = (const unsigned char*)d_in[2];
    const float*         value      = (const float*)d_in[3];
    const float*         next_value = (const float*)d_in[4];
    const float*         discount   = (const float*)d_in[5];

    const int B = in_sizes[0] / T_DIM;            // 8192
    float* out  = (float*)d_out;
    float* advp = out;                            // advantages [B, T]
    float* retp = out + (size_t)B * T_DIM;        // returns    [B, T]

    gae_scan_kernel<<<B, BLK, 0, stream>>>(reward, terminated, done,
                                           value, next_value, discount,
                                           advp, retp);
}